// LocalNetwork_40999757807945
// MI455X (gfx1250) — compile-verified
//
#include <hip/hip_runtime.h>

// STN forward for MI455X (gfx1250, wave32).
// GEMM1 (128 x 150528 @ 150528 x 20) is memory bound: 89 MB / 23.3 TB/s ~ 4us.
// fp32 WMMA 16x16x4 split-K; 8 waves/WG (one M-tile each); double-buffered
// software pipeline so the next block's 12 b64 loads are in flight while the
// current block's 8 WMMAs execute.

typedef __attribute__((ext_vector_type(2))) float v2f;
typedef __attribute__((ext_vector_type(8))) float v8f;

#define NB    128
#define CH    3
#define HH    224
#define WW    224
#define KDIM  150528          // CH*HH*WW
#define KSTEPS_TOTAL 37632    // KDIM / 4
#define NP    32              // hidden dim 20 padded to 2 WMMA N-tiles

struct Blk { v2f a[4], b0[4], b1[4]; };

__device__ __forceinline__ void load_blk(Blk& B, const float* pA,
                                         const float* pB0, const float* pB1,
                                         long s, int hi, float mask1)
{
#pragma unroll
  for (int u = 0; u < 4; ++u) {
    const long col = 4 * (s + u) + 2 * hi;       // even -> aligned b64 loads
    B.a[u]  = *(const v2f*)(pA  + col);
    B.b0[u] = *(const v2f*)(pB0 + col);
    B.b1[u] = *(const v2f*)(pB1 + col) * mask1;  // branchless N-pad mask
  }
}

__device__ __forceinline__ void mma_blk(const Blk& B, v8f& acc0, v8f& acc1)
{
#pragma unroll
  for (int u = 0; u < 4; ++u) {
    acc0 = __builtin_amdgcn_wmma_f32_16x16x4_f32(
        false, B.a[u], false, B.b0[u], (short)0, acc0, false, false);
    acc1 = __builtin_amdgcn_wmma_f32_16x16x4_f32(
        false, B.a[u], false, B.b1[u], (short)0, acc1, false, false);
  }
}

// ---------------------------------------------------------------------------
// Kernel 1: split-K GEMM  part[chunk][128][32] = x_chunk @ W1.T_chunk
// 256 threads = 8 waves; wave w owns M-tile w (rows 16w..16w+15), both N-tiles.
// A (16x4 f32) layout: lanes 0-15 hold K=0,1 in v[0],v[1]; lanes 16-31 K=2,3.
// B (4x16 f32) mirrors A with N striped across lanes.
// ---------------------------------------------------------------------------
__global__ __launch_bounds__(256) void stn_gemm1(
    const float* __restrict__ img, const float* __restrict__ W1,
    float* __restrict__ part, int ksteps_per_chunk)
{
  const int tid  = threadIdx.x;
  const int wave = tid >> 5;        // 0..7 -> M-tile
  const int lane = tid & 31;
  const int lo   = lane & 15;
  const int hi   = lane >> 4;
  const int chunk = blockIdx.x;

  long s0 = (long)chunk * ksteps_per_chunk;
  long s1 = s0 + ksteps_per_chunk;
  if (s1 > KSTEPS_TOTAL) s1 = KSTEPS_TOTAL;

  const float* pA  = img + (long)(wave * 16 + lo) * KDIM;
  const float* pB0 = W1 + (long)lo * KDIM;
  const int row1c  = 16 + (lo < 4 ? lo : 3);     // clamp into valid rows 16..19
  const float* pB1 = W1 + (long)row1c * KDIM;
  const float mask1 = (lo < 4) ? 1.f : 0.f;      // zero padded N cols 20..31

  const v8f vz = {0.f,0.f,0.f,0.f,0.f,0.f,0.f,0.f};
  v8f acc0 = vz, acc1 = vz;

  const long nblk = (s1 > s0) ? ((s1 - s0) >> 2) : 0;  // 4-step blocks
  const long srem = s0 + (nblk << 2);                  // scalar tail start

  if (nblk > 0) {
    Blk B0, B1;
    load_blk(B0, pA, pB0, pB1, s0, hi, mask1);         // prologue
    long b = 0;
    for (; b + 2 < nblk; b += 2) {                     // steady state (x2)
      load_blk(B1, pA, pB0, pB1, s0 + 4 * (b + 1), hi, mask1);
      mma_blk(B0, acc0, acc1);
      load_blk(B0, pA, pB0, pB1, s0 + 4 * (b + 2), hi, mask1);
      mma_blk(B1, acc0, acc1);
    }
    if (b + 2 == nblk) {                               // two blocks left
      load_blk(B1, pA, pB0, pB1, s0 + 4 * (b + 1), hi, mask1);
      mma_blk(B0, acc0, acc1);
      mma_blk(B1, acc0, acc1);
    } else {                                           // one block left
      mma_blk(B0, acc0, acc1);
    }
  }
  for (long s = srem; s < s1; ++s) {                   // scalar tail
    const long col = 4 * s + 2 * hi;
    v2f a  = *(const v2f*)(pA  + col);
    v2f b0 = *(const v2f*)(pB0 + col);
    v2f b1 = *(const v2f*)(pB1 + col) * mask1;
    acc0 = __builtin_amdgcn_wmma_f32_16x16x4_f32(
        false, a, false, b0, (short)0, acc0, false, false);
    acc1 = __builtin_amdgcn_wmma_f32_16x16x4_f32(
        false, a, false, b1, (short)0, acc1, false, false);
  }

  // C/D layout: VGPR r -> lanes 0-15: (M=r, N=lane); lanes 16-31: (M=r+8).
  // Always store (zeros for empty chunks) since ws is poisoned.
  float* op = part + (long)chunk * (NB * NP);
#pragma unroll
  for (int r = 0; r < 8; ++r) {
    const int row = wave * 16 + r + 8 * hi;
    op[row * NP + lo]      = acc0[r];
    op[row * NP + 16 + lo] = acc1[r];
  }
}

// ---------------------------------------------------------------------------
// Kernel 2: reduce partials -> h1 = tanh(.+b1), theta = tanh(h1@W2.T + b2)
// ---------------------------------------------------------------------------
__global__ __launch_bounds__(32) void stn_head(
    const float* __restrict__ part, int nchunk,
    const float* __restrict__ b1, const float* __restrict__ W2,
    const float* __restrict__ b2, float* __restrict__ theta)
{
  __shared__ float sh[20];
  const int n = blockIdx.x;
  const int t = threadIdx.x;
  if (t < 20) {
    float s = b1[t];
    const float* p = part + n * NP + t;
    for (int c = 0; c < nchunk; ++c) s += p[(long)c * (NB * NP)];
    sh[t] = tanhf(s);
  }
  __syncthreads();
  if (t < 6) {
    float s = b2[t];
#pragma unroll
    for (int j = 0; j < 20; ++j) s += W2[t * 20 + j] * sh[j];
    theta[n * 6 + t] = tanhf(s);
  }
}

// ---------------------------------------------------------------------------
// Kernel 3: affine grid + bilinear gather. One thread per (n,y,x), 3 channels.
// ---------------------------------------------------------------------------
__global__ __launch_bounds__(256) void stn_sample(
    const float* __restrict__ img, const float* __restrict__ theta,
    float* __restrict__ out)
{
  const int HW = HH * WW;
  int idx = blockIdx.x * 256 + threadIdx.x;
  if (idx >= NB * HW) return;
  const int x = idx % WW;
  const int y = (idx / WW) % HH;
  const int n = idx / HW;

  const float* th = theta + n * 6;
  const float t00 = th[0], t01 = th[1], t02 = th[2];
  const float t10 = th[3], t11 = th[4], t12 = th[5];

  const float xs = (2.f * (float)x + 1.f) * (1.f / (float)WW) - 1.f;
  const float ys = (2.f * (float)y + 1.f) * (1.f / (float)HH) - 1.f;
  const float gx = t00 * xs + t01 * ys + t02;
  const float gy = t10 * xs + t11 * ys + t12;
  const float ix = ((gx + 1.f) * (float)WW - 1.f) * 0.5f;
  const float iy = ((gy + 1.f) * (float)HH - 1.f) * 0.5f;

  const float x0 = floorf(ix), y0 = floorf(iy);
  const float wx1 = ix - x0, wx0 = 1.f - wx1;
  const float wy1 = iy - y0, wy0 = 1.f - wy1;

  float r0 = 0.f, r1 = 0.f, r2 = 0.f;
  const float* base = img + (long)n * CH * HW;

  auto tap = [&](float yc, float xc, float wgt) {
    bool v = (xc >= 0.f) && (xc <= (float)(WW - 1)) &&
             (yc >= 0.f) && (yc <= (float)(HH - 1));
    float m = v ? wgt : 0.f;  // branchless: clamped loads are always in-range
    int xi = (int)fminf(fmaxf(xc, 0.f), (float)(WW - 1));
    int yi = (int)fminf(fmaxf(yc, 0.f), (float)(HH - 1));
    long o = (long)yi * WW + xi;
    r0 += m * base[o];
    r1 += m * base[o + HW];
    r2 += m * base[o + 2 * HW];
  };
  tap(y0,       x0,       wx0 * wy0);
  tap(y0,       x0 + 1.f, wx1 * wy0);
  tap(y0 + 1.f, x0,       wx0 * wy1);
  tap(y0 + 1.f, x0 + 1.f, wx1 * wy1);

  long o = (long)n * CH * HW + (long)y * WW + x;
  out[o] = r0;
  out[o + HW] = r1;
  out[o + 2 * HW] = r2;
}

// ---------------------------------------------------------------------------
extern "C" void kernel_launch(void* const* d_in, const int* in_sizes, int n_in,
                              void* d_out, int out_size, void* d_ws, size_t ws_size,
                              hipStream_t stream)
{
  const float* img = (const float*)d_in[0];
  const float* W1  = (const float*)d_in[1];
  const float* b1  = (const float*)d_in[2];
  const float* W2  = (const float*)d_in[3];
  const float* b2  = (const float*)d_in[4];
  float* out = (float*)d_out;

  // Split-K: 588 chunks of 256 K-values -> 4704 waves (adapt down to ws_size).
  const size_t theta_bytes = NB * 6 * sizeof(float);
  int nchunk = 588;
  while (nchunk > 1 &&
         (size_t)nchunk * NB * NP * sizeof(float) + theta_bytes > ws_size)
    nchunk >>= 1;
  const int ksteps = (KSTEPS_TOTAL + nchunk - 1) / nchunk;

  float* part  = (float*)d_ws;
  float* theta = part + (size_t)nchunk * NB * NP;

  stn_gemm1<<<dim3(nchunk), dim3(256), 0, stream>>>(img, W1, part, ksteps);
  stn_head<<<dim3(NB), dim3(32), 0, stream>>>(part, nchunk, b1, W2, b2, theta);
  const int total = NB * HH * WW;
  stn_sample<<<dim3((total + 255) / 256), dim3(256), 0, stream>>>(img, theta, out);
}